// MinibatchDiscrimination_89189290868771
// MI455X (gfx1250) — compile-verified
//
#include <hip/hip_runtime.h>
#include <hip/hip_bf16.h>

typedef __attribute__((ext_vector_type(2))) float v2f;
typedef __attribute__((ext_vector_type(8))) float v8f;

#define BSZ 512      // batch
#define FSZ 512      // features (GEMM K-dim)
#define KSZ 128      // minibatch kernels
#define DSZ 16       // kernel depth
#define ND  (KSZ * DSZ)   // 2048, GEMM N-dim
#define OUTW (FSZ + KSZ)  // 640

// ---------------------------------------------------------------------------
// Stage 1: m[b, k*16+d] = sum_f x[b,f] * w[f, k*16+d]
// fp32 WMMA 16x16x4. Block = 256 threads = 8 wave32; all waves share one
// 16-row A tile staged in LDS (padded rows: stride 516 dwords -> the 16
// lanes of a half-wave hit distinct banks on the A-fragment ds_load_b64).
// grid = (32 tilesM, 16 groups of 8 tilesN)
// ---------------------------------------------------------------------------
__global__ __launch_bounds__(256)
void mbdisc_gemm_wmma(const float* __restrict__ X,
                      const float* __restrict__ W,
                      float* __restrict__ M) {
    constexpr int APAD = 516;                 // 512 + 4 pad (bank spread)
    __shared__ float lds_a[16 * APAD];        // ~33 KB

    const int tid     = threadIdx.x;
    const int rowBase = blockIdx.x * 16;      // tileM * 16

    // Cooperative, coalesced load of A tile (16 x 512 fp32)
    for (int i = tid; i < 16 * FSZ; i += 256) {
        const int r = i >> 9;          // /512
        const int c = i & (FSZ - 1);
        lds_a[r * APAD + c] = X[(rowBase + r) * FSZ + c];
    }
    __syncthreads();

    const int wave = tid >> 5;                // 0..7 (wave32)
    const int lane = tid & 31;
    const int half = lane >> 4;               // 0: lanes 0-15, 1: lanes 16-31
    const int l16  = lane & 15;
    const int n    = (blockIdx.y * 8 + wave) * 16 + l16;  // global N column
    const int kb   = half * 2;                // K offset within 16x16x4 step

    v8f acc = {};
    for (int k = 0; k < FSZ; k += 4) {
        // A fragment (ISA 32-bit A 16x4): lane l16 = row M, vgpr pair = K
        v2f a = *(const v2f*)&lds_a[l16 * APAD + k + kb];
        // B fragment: lane l16 = column N, vgpr pair = K (row-major W rows
        // -> 64B contiguous per half-wave)
        v2f b;
        b.x = W[(k + kb)     * ND + n];
        b.y = W[(k + kb + 1) * ND + n];
        if (k + 4 < FSZ)
            __builtin_prefetch(&W[(k + 4) * ND + n], 0, 3);  // global_prefetch_b8
        // (neg_a, A, neg_b, B, c_mod, C, reuse_a, reuse_b)
        acc = __builtin_amdgcn_wmma_f32_16x16x4_f32(
            false, a, false, b, (short)0, acc, false, false);
    }

    // C/D layout: VGPR v -> M = v + 8*half, N = l16 (within tile)
    const int crow = rowBase + half * 8;
#pragma unroll
    for (int v = 0; v < 8; ++v) {
        M[(crow + v) * ND + n] = acc[v];
    }
}

// ---------------------------------------------------------------------------
// Stage 2: mb_feats[b,k] = sum_{b'} exp(-sum_d |m[b,k,d]-m[b',k,d]|)
// One block per (k, 256-row b tile). Full m[:,k,:] slice (512x16 fp32,
// 32 KB) staged in LDS; inner b' loop is a same-address LDS broadcast.
// grid = (128 k, 2 b-tiles), block = 256
// ---------------------------------------------------------------------------
__global__ __launch_bounds__(256)
void mbdisc_pairwise(const float* __restrict__ M,
                     float* __restrict__ out) {
    __shared__ __align__(16) float lds[BSZ * DSZ];   // 32 KB

    const int k   = blockIdx.x;
    const int tid = threadIdx.x;

    // Load m[:, k, :] as float4s (rows are 64B-aligned: k*16 floats offset)
    const float* src = M + k * DSZ;
    for (int i = tid; i < BSZ * 4; i += 256) {        // 2048 float4s
        const int b  = i >> 2;
        const int d4 = i & 3;
        *(float4*)&lds[b * DSZ + d4 * 4] =
            *(const float4*)&src[b * ND + d4 * 4];
    }
    __syncthreads();

    const int b = blockIdx.y * 256 + tid;
    float my[DSZ];
#pragma unroll
    for (int d = 0; d < DSZ; ++d) my[d] = lds[b * DSZ + d];

    float sum = 0.0f;
    for (int bp = 0; bp < BSZ; ++bp) {
        const float* row = &lds[bp * DSZ];
        float l1 = 0.0f;
#pragma unroll
        for (int d = 0; d < DSZ; ++d) l1 += fabsf(my[d] - row[d]);
        sum += __expf(-l1);                 // v_exp_f32
    }
    out[b * OUTW + FSZ + k] = sum;
}

// ---------------------------------------------------------------------------
// Stage 3: out[b, 0:512] = x[b, :]  (float4 copy; 640-float rows keep 16B
// alignment: 640*4 % 16 == 0)
// ---------------------------------------------------------------------------
__global__ __launch_bounds__(256)
void mbdisc_copy_x(const float* __restrict__ X, float* __restrict__ out) {
    const int i = blockIdx.x * blockDim.x + threadIdx.x;  // over 512*128 f4
    const int b  = i >> 7;          // /128 float4s per row
    const int f4 = i & 127;
    *(float4*)&out[b * OUTW + f4 * 4] = *(const float4*)&X[b * FSZ + f4 * 4];
}

extern "C" void kernel_launch(void* const* d_in, const int* in_sizes, int n_in,
                              void* d_out, int out_size, void* d_ws, size_t ws_size,
                              hipStream_t stream) {
    const float* x = (const float*)d_in[0];   // [512, 512]
    const float* w = (const float*)d_in[1];   // [512, 128, 16] row-major = [512, 2048]
    float* out = (float*)d_out;               // [512, 640]
    float* m   = (float*)d_ws;                // [512, 2048] scratch (4 MB)

    mbdisc_gemm_wmma<<<dim3(32, 16), 256, 0, stream>>>(x, w, m);
    mbdisc_pairwise<<<dim3(128, 2), 256, 0, stream>>>(m, out);
    mbdisc_copy_x<<<dim3(256), 256, 0, stream>>>(x, out);
}